// IndexMasking_42623255446178
// MI455X (gfx1250) — compile-verified
//
#include <hip/hip_runtime.h>
#include <stdint.h>

// Problem constants (from reference setup_inputs)
#define B_    64
#define L_    550
#define D_    768
#define R_    3
#define P_    50
#define KEEP_ 400          // L - R*P
#define NSORT 1024         // pow2 padding for bitonic sort

#ifndef __has_builtin
#define __has_builtin(x) 0
#endif

// Builtin signature (verified by round-1 diagnostics on this toolchain):
//   void __builtin_amdgcn_global_load_async_to_lds_b128(
//       v4i addrspace(1)* gsrc, v4i addrspace(3)* ldst, imm offset, imm cpol)
//   void __builtin_amdgcn_global_store_async_from_lds_b128(
//       v4i addrspace(1)* gdst, v4i addrspace(3)* lsrc, imm offset, imm cpol)
typedef int v4i __attribute__((vector_size(16)));
typedef __attribute__((address_space(1))) v4i* g_v4i_p;
typedef __attribute__((address_space(3))) v4i* l_v4i_p;

#if __has_builtin(__builtin_amdgcn_global_load_async_to_lds_b128) && \
    __has_builtin(__builtin_amdgcn_global_store_async_from_lds_b128)
#define HAVE_ASYNC_LDS 1
#else
#define HAVE_ASYNC_LDS 0
#endif

__device__ __forceinline__ void wait_async0() {
#if __has_builtin(__builtin_amdgcn_s_wait_asynccnt)
  __builtin_amdgcn_s_wait_asynccnt(0);
#else
  asm volatile("s_wait_asynccnt 0" ::: "memory");
#endif
}

// ---------------------------------------------------------------------------
// Kernel A (emitted first so the async ops show in the disasm snippet):
// row gather x_masked[row] = x[b, ids_keep[row], :].
// 4 rows (4*3KB = 12KB LDS) per 256-thread block; each lane moves one 16B
// chunk per iteration via the CDNA5 async global->LDS->global path
// (ASYNCcnt-tracked, bypasses the VGPR file). Each lane stores exactly the
// LDS bytes it loaded, so a single s_wait_asynccnt 0 between the load and
// store phases suffices — no cross-lane barrier needed.
// Traffic: 78.6 MB in + 78.6 MB out -> ~6.8 us at 23.3 TB/s HBM.
// ---------------------------------------------------------------------------
__global__ __launch_bounds__(256) void IndexMasking_gather_kernel(
    const float* __restrict__ x, const int* __restrict__ ids_keep,
    float* __restrict__ xm)
{
  const int tid      = threadIdx.x;
  const int base_row = blockIdx.x * 4;       // rows of the packed output

#if HAVE_ASYNC_LDS
  __shared__ __align__(16) float staged[4 * D_];

  #pragma unroll
  for (int i = 0; i < 3; ++i) {
    int linear = i * 256 + tid;              // float4 slot 0..767
    int lrow   = linear / 192;               // row within block (D_/4 = 192)
    int c4     = linear % 192;
    int row    = base_row + lrow;
    int b      = row / KEEP_;
    int k      = ids_keep[row];
    const float* src = x + ((size_t)(b * L_ + k) * D_ + (size_t)c4 * 4);
    __builtin_amdgcn_global_load_async_to_lds_b128(
        (g_v4i_p)(uintptr_t)src, (l_v4i_p)(staged + (size_t)linear * 4), 0, 0);
  }
  wait_async0();  // LDS now holds all 12KB

  #pragma unroll
  for (int i = 0; i < 3; ++i) {
    int linear = i * 256 + tid;
    float* dst = xm + ((size_t)base_row * D_ + (size_t)linear * 4);
    __builtin_amdgcn_global_store_async_from_lds_b128(
        (g_v4i_p)(uintptr_t)dst, (l_v4i_p)(staged + (size_t)linear * 4), 0, 0);
  }
  wait_async0();  // stores complete before wave retires
#else
  // Fallback: straight register copy (global_load_b128 / global_store_b128).
  #pragma unroll
  for (int i = 0; i < 3; ++i) {
    int linear = i * 256 + tid;
    int lrow   = linear / 192;
    int c4     = linear % 192;
    int row    = base_row + lrow;
    int b      = row / KEEP_;
    int k      = ids_keep[row];
    const float4* src =
        (const float4*)(x + ((size_t)(b * L_ + k) * D_ + (size_t)c4 * 4));
    float4* dst = (float4*)(xm + ((size_t)base_row * D_ + (size_t)linear * 4));
    *dst = *src;
  }
#endif
}

// ---------------------------------------------------------------------------
// Kernel B: per-batch-row masked noise -> stable argsort -> ids/mask outputs.
// One 256-thread (8-wave) block per batch row; 8KB LDS of 64-bit sort keys.
// Key = (float_bits(noise) << 16) | index : noise >= 0 so IEEE bit pattern is
// monotonic, and the index in the low bits reproduces jnp.argsort's stable
// tie-breaking exactly (the 150 entries forced to 2.0 tie).
// ---------------------------------------------------------------------------
__global__ __launch_bounds__(256) void IndexMasking_sort_kernel(
    const float* __restrict__ noise, const int* __restrict__ indexes,
    float* __restrict__ mask_out, float* __restrict__ idsr_out,
    int* __restrict__ ids_keep)
{
  __shared__ uint64_t keys[NSORT];
  const int b   = blockIdx.x;
  const int tid = threadIdx.x;

  // Fixed 4-iteration fill: NSORT / 256 == 4.
  #pragma unroll
  for (int i = 0; i < 4; ++i) {
    int l = i * 256 + tid;
    uint64_t kk = ~0ull;                       // pad: sorts to the end
    if (l < L_) {
      uint32_t fb = __float_as_uint(noise[b * L_ + l]);
      kk = ((uint64_t)fb << 16) | (uint32_t)l;
    }
    keys[l] = kk;
  }
  __syncthreads();

  // noise.at[b, pos].set(2.0): 150 positions; duplicate indexes write
  // identical 64-bit values so the race is benign.
  if (tid < R_ * P_) {
    int r = tid / P_, j = tid % P_;
    int pos = indexes[b * R_ + r] * P_ + j;              // < 550 by construction
    keys[pos] = ((uint64_t)0x40000000u << 16) | (uint32_t)pos;  // bits of 2.0f
  }
  __syncthreads();

  // Bitonic sort, 1024 elements, 256 threads (4 disjoint pairs per thread).
  for (unsigned k = 2; k <= NSORT; k <<= 1) {
    for (unsigned j = k >> 1; j > 0; j >>= 1) {
      #pragma unroll
      for (int t = 0; t < 4; ++t) {
        unsigned idx = (unsigned)(t * 256 + tid);
        unsigned ixj = idx ^ j;
        if (ixj > idx) {
          uint64_t a = keys[idx], c = keys[ixj];
          bool up = ((idx & k) == 0);
          if ((a > c) == up) { keys[idx] = c; keys[ixj] = a; }
        }
      }
      __syncthreads();
    }
  }

  // keys sorted ascending; rank i holds original index s = ids_shuffle[i].
  // ids_restore[s] = i ; mask[s] = (i >= 400) ; ids_keep[i] = s for i < 400.
  for (int i = tid; i < L_; i += 256) {
    int s = (int)(keys[i] & 0xFFFFu);
    idsr_out[b * L_ + s] = (float)i;                 // d_out is float
    mask_out[b * L_ + s] = (i < KEEP_) ? 0.0f : 1.0f;
    if (i < KEEP_) ids_keep[b * KEEP_ + i] = s;
  }
}

// ---------------------------------------------------------------------------
// d_in:  [0] x (B,L,D) f32   [1] noise (B,L) f32   [2] indexes (B,R) i32
// d_out: x_masked (B,400,D) f32 | mask (B,L) f32 | ids_restore (B,L) as f32
// d_ws:  ids_keep (B,400) i32  = 100KB
// ---------------------------------------------------------------------------
extern "C" void kernel_launch(void* const* d_in, const int* in_sizes, int n_in,
                              void* d_out, int out_size, void* d_ws, size_t ws_size,
                              hipStream_t stream) {
  (void)in_sizes; (void)n_in; (void)out_size; (void)ws_size;
  const float* x       = (const float*)d_in[0];
  const float* noise   = (const float*)d_in[1];
  const int*   indexes = (const int*)d_in[2];

  float* xm   = (float*)d_out;                      // 64*400*768
  float* mask = xm + (size_t)B_ * KEEP_ * D_;       // 64*550
  float* idsr = mask + (size_t)B_ * L_;             // 64*550
  int*   ids_keep = (int*)d_ws;                     // 64*400 ints

  IndexMasking_sort_kernel<<<B_, 256, 0, stream>>>(noise, indexes, mask, idsr,
                                                   ids_keep);
  IndexMasking_gather_kernel<<<(B_ * KEEP_) / 4, 256, 0, stream>>>(x, ids_keep,
                                                                   xm);
}